// Mamba2Mixer_9491877724665
// MI455X (gfx1250) — compile-verified
//
#include <hip/hip_runtime.h>
#include <math.h>

#define HID       2560
#define NH        80
#define PDIM      64
#define NSTATE    128
#define KCONV     4
#define INTER     (NH * PDIM)                 // 5120
#define CONV_DIM  (INTER + 2 * NSTATE)        // 5376 (G=1)
#define PROJ_DIM  (INTER + CONV_DIM + NH)     // 10576
#define BATCH     64
#define EPS       1e-5f

typedef float v2f __attribute__((ext_vector_type(2)));
typedef float v4f __attribute__((ext_vector_type(4)));
typedef float v8f __attribute__((ext_vector_type(8)));

__device__ __forceinline__ float silu_f(float x) {
    return x / (1.0f + __expf(-x));
}

__device__ __forceinline__ v2f lo2(v4f v) { return __builtin_shufflevector(v, v, 0, 1); }
__device__ __forceinline__ v2f hi2(v4f v) { return __builtin_shufflevector(v, v, 2, 3); }

// One 16-K group of fragments: B row slice + 4 A-tile row slices, 8 floats each.
struct Frag {
    v4f b0, b1;
    v4f a00, a01, a10, a11, a20, a21, a30, a31;
};

// ---------------------------------------------------------------------------
// C[64, N] (+)= A[64, K] * W[N, K]^T   using V_WMMA_F32_16X16X4_F32.
// One wave owns one 16-wide N tile and all four 16-row M tiles (M=64),
// split-K across `nKSplit` waves with global fp32 atomics (C pre-zeroed).
//
// K remap (sum over K is commutative, A/B just must agree): per 16-K group,
// lanes 0-15 hold logical K = base+0..7 (contiguous), lanes 16-31 hold
// K = base+8..15; WMMA step u consumes elements {2u,2u+1} of each lane's 8.
// => per row per group: two global_load_b128 (instead of 4x b64), and the
// lo2/hi2 slices are even-aligned VGPR pairs consumed directly by WMMA.
//
// Double-buffered pipeline (manual x2 unroll, no copies): issue the 10 b128
// loads of the next group, then run 16 WMMAs on the current group, so loads
// stay in flight underneath the matrix ops.
//
// D (16x16 f32): vgpr i, lane l -> M = i + 8*(l>>4), Nc = l&15.
// Requires kchunk % 32 == 0 (kchunk = 640 for both GEMMs here).
// ---------------------------------------------------------------------------
__global__ __launch_bounds__(256) void gemm_nt_f32_wmma(
    const float* __restrict__ A, const float* __restrict__ W,
    float* __restrict__ C, int N, int K, int nKSplit)
{
    const int wave   = blockIdx.x * (blockDim.x >> 5) + (threadIdx.x >> 5);
    const int lane   = threadIdx.x & 31;
    const int nTiles = N >> 4;
    const int jt     = wave / nKSplit;     // N tile
    const int ks     = wave % nKSplit;     // K split
    if (jt >= nTiles) return;              // wave-uniform: EXEC stays all-1s

    const int nbase  = jt << 4;
    const int kchunk = K / nKSplit;
    const int k0     = ks * kchunk;

    const int r  = lane & 15;              // M row (A) / N row-of-W (B)
    const int kh = (lane >> 4) * 8;        // contiguous K half: 0 or 8

    const float* pb  = W + (size_t)(nbase + r) * K + k0 + kh;
    const float* pa0 = A + (size_t)r * K + k0 + kh;
    const float* pa1 = pa0 + (size_t)16 * K;
    const float* pa2 = pa0 + (size_t)32 * K;
    const float* pa3 = pa0 + (size_t)48 * K;

    v8f acc0 = {0,0,0,0,0,0,0,0};
    v8f acc1 = acc0, acc2 = acc0, acc3 = acc0;

#define LOADG(F) do {                                               \
        (F).b0  = *(const v4f*)(pb);   (F).b1  = *(const v4f*)(pb + 4);   \
        (F).a00 = *(const v4f*)(pa0);  (F).a01 = *(const v4f*)(pa0 + 4);  \
        (F).a10 = *(const v4f*)(pa1);  (F).a11 = *(const v4f*)(pa1 + 4);  \
        (F).a20 = *(const v4f*)(pa2);  (F).a21 = *(const v4f*)(pa2 + 4);  \
        (F).a30 = *(const v4f*)(pa3);  (F).a31 = *(const v4f*)(pa3 + 4);  \
        pb += 16; pa0 += 16; pa1 += 16; pa2 += 16; pa3 += 16;             \
    } while (0)

#define WMMA4(A0, A1, A2, A3, BS) do {                                                                  \
        acc0 = __builtin_amdgcn_wmma_f32_16x16x4_f32(false, (A0), false, (BS), (short)0, acc0, false, false); \
        acc1 = __builtin_amdgcn_wmma_f32_16x16x4_f32(false, (A1), false, (BS), (short)0, acc1, false, false); \
        acc2 = __builtin_amdgcn_wmma_f32_16x16x4_f32(false, (A2), false, (BS), (short)0, acc2, false, false); \
        acc3 = __builtin_amdgcn_wmma_f32_16x16x4_f32(false, (A3), false, (BS), (short)0, acc3, false, false); \
    } while (0)

#define COMPUTE(F) do {                                                                   \
        WMMA4(lo2((F).a00), lo2((F).a10), lo2((F).a20), lo2((F).a30), lo2((F).b0));       \
        WMMA4(hi2((F).a00), hi2((F).a10), hi2((F).a20), hi2((F).a30), hi2((F).b0));       \
        WMMA4(lo2((F).a01), lo2((F).a11), lo2((F).a21), lo2((F).a31), lo2((F).b1));       \
        WMMA4(hi2((F).a01), hi2((F).a11), hi2((F).a21), hi2((F).a31), hi2((F).b1));       \
    } while (0)

    const int groups = kchunk >> 4;        // 16 K per group; 40 here (even)
    Frag fA, fB;
    LOADG(fA);
    for (int it = 0; it < (groups >> 1) - 1; ++it) {
        LOADG(fB);
        COMPUTE(fA);
        LOADG(fA);
        COMPUTE(fB);
    }
    LOADG(fB);
    COMPUTE(fA);
    COMPUTE(fB);

#undef LOADG
#undef WMMA4
#undef COMPUTE

    const int col = nbase + (lane & 15);
    const int rhi = (lane >> 4) * 8;
    #pragma unroll
    for (int i = 0; i < 8; ++i) {
        atomicAdd(&C[(size_t)( 0 + rhi + i) * N + col], acc0[i]);
        atomicAdd(&C[(size_t)(16 + rhi + i) * N + col], acc1[i]);
        atomicAdd(&C[(size_t)(32 + rhi + i) * N + col], acc2[i]);
        atomicAdd(&C[(size_t)(48 + rhi + i) * N + col], acc3[i]);
    }
}

// ---------------------------------------------------------------------------
// Per-batch: causal-conv update + SiLU, split into hs / Bm / Cm,
// BC = <Bm,Cm> (G=1 -> scalar), dt softplus/clip, dA = exp(dt * -exp(A_log)).
// ---------------------------------------------------------------------------
__global__ __launch_bounds__(256) void conv_dt_kernel(
    const float* __restrict__ proj, const float* __restrict__ conv_state,
    const float* __restrict__ conv_w, const float* __restrict__ conv_b,
    const float* __restrict__ dt_bias, const float* __restrict__ A_log,
    float* __restrict__ hs_ws, float* __restrict__ cm_ws,
    float* __restrict__ dt_ws, float* __restrict__ da_ws, float* __restrict__ bc_ws)
{
    const int b   = blockIdx.x;
    const int tid = threadIdx.x;
    __shared__ float sBC[2 * NSTATE];
    __shared__ float red[256];

    const float* projb = proj + (size_t)b * PROJ_DIM;

    for (int c = tid; c < CONV_DIM; c += 256) {
        const float4 cs = *(const float4*)(conv_state + ((size_t)b * CONV_DIM + c) * KCONV);
        const float4 w  = *(const float4*)(conv_w + (size_t)c * KCONV);
        const float hbc = projb[INTER + c];
        float x = cs.y * w.x + cs.z * w.y + cs.w * w.z + hbc * w.w + conv_b[c];
        x = silu_f(x);
        if (c < INTER) {
            hs_ws[(size_t)b * INTER + c] = x;
        } else {
            sBC[c - INTER] = x;
            if (c >= INTER + NSTATE)
                cm_ws[(size_t)b * NSTATE + (c - INTER - NSTATE)] = x;
        }
    }
    __syncthreads();

    float p = (tid < NSTATE) ? sBC[tid] * sBC[tid + NSTATE] : 0.0f;
    red[tid] = p;
    __syncthreads();
    for (int s = 128; s > 0; s >>= 1) {
        if (tid < s) red[tid] += red[tid + s];
        __syncthreads();
    }
    if (tid == 0) bc_ws[b] = red[0];

    if (tid < NH) {
        const float dtv = projb[INTER + CONV_DIM + tid] + dt_bias[tid];
        float sp = (dtv > 20.0f) ? dtv : log1pf(__expf(dtv));
        sp = fminf(fmaxf(sp, 0.0f), 100.0f);
        dt_ws[(size_t)b * NH + tid] = sp;
        da_ws[(size_t)b * NH + tid] = __expf(sp * (-__expf(A_log[tid])));
    }
}

// ---------------------------------------------------------------------------
// SSM contraction. Block = (head h, batch b), 8 waves x 8 rows.
// Each row: one float4/lane coalesced read of ssm_state (168 MB stream),
// dot with Cm via wave32 shfl_xor reduction.
// y = dA * <S_row, Cm> + hs * (dt*BC + D)
// ---------------------------------------------------------------------------
__global__ __launch_bounds__(256) void ssm_kernel(
    const float* __restrict__ ssm_state, const float* __restrict__ cm_ws,
    const float* __restrict__ hs_ws, const float* __restrict__ dt_ws,
    const float* __restrict__ da_ws, const float* __restrict__ bc_ws,
    const float* __restrict__ Dvec, float* __restrict__ y_ws)
{
    const int h    = blockIdx.x;
    const int b    = blockIdx.y;
    const int lane = threadIdx.x & 31;
    const int wv   = threadIdx.x >> 5;

    const float4 c4  = *(const float4*)(cm_ws + (size_t)b * NSTATE + lane * 4);
    const float dA   = da_ws[(size_t)b * NH + h];
    const float coef = dt_ws[(size_t)b * NH + h] * bc_ws[b] + Dvec[h];
    const float* base = ssm_state + ((size_t)b * NH + h) * (size_t)PDIM * NSTATE;

    #pragma unroll
    for (int i = 0; i < 8; ++i) {
        const int p = wv * 8 + i;
        const float4 s4 = *(const float4*)(base + (size_t)p * NSTATE + lane * 4);
        float d = s4.x * c4.x + s4.y * c4.y + s4.z * c4.z + s4.w * c4.w;
        d += __shfl_xor(d, 16);
        d += __shfl_xor(d, 8);
        d += __shfl_xor(d, 4);
        d += __shfl_xor(d, 2);
        d += __shfl_xor(d, 1);
        if (lane == 0) {
            const size_t idx = (size_t)b * INTER + h * PDIM + p;
            y_ws[idx] = dA * d + hs_ws[idx] * coef;
        }
    }
}

// ---------------------------------------------------------------------------
// Per-batch: yf = y * silu(gate); RMS-norm over INTER; yn = norm_w * yf * rsqrt.
// ---------------------------------------------------------------------------
__global__ __launch_bounds__(256) void gatenorm_kernel(
    const float* __restrict__ proj, const float* __restrict__ y_ws,
    const float* __restrict__ norm_w, float* __restrict__ yn_ws)
{
    const int b   = blockIdx.x;
    const int tid = threadIdx.x;
    __shared__ float yf[INTER];      // 20 KB
    __shared__ float red[256];

    const float* gate = proj + (size_t)b * PROJ_DIM;
    float ss = 0.0f;
    for (int i = tid; i < INTER; i += 256) {
        const float v = y_ws[(size_t)b * INTER + i] * silu_f(gate[i]);
        yf[i] = v;
        ss += v * v;
    }
    red[tid] = ss;
    __syncthreads();
    for (int s = 128; s > 0; s >>= 1) {
        if (tid < s) red[tid] += red[tid + s];
        __syncthreads();
    }
    const float inv = rsqrtf(red[0] / (float)INTER + EPS);
    for (int i = tid; i < INTER; i += 256)
        yn_ws[(size_t)b * INTER + i] = norm_w[i] * yf[i] * inv;
}

// ---------------------------------------------------------------------------
extern "C" void kernel_launch(void* const* d_in, const int* in_sizes, int n_in,
                              void* d_out, int out_size, void* d_ws, size_t ws_size,
                              hipStream_t stream)
{
    const float* input_states = (const float*)d_in[0];
    const float* conv_state   = (const float*)d_in[1];
    const float* ssm_state    = (const float*)d_in[2];
    const float* in_proj_w    = (const float*)d_in[3];
    const float* conv_w       = (const float*)d_in[4];
    const float* conv_b       = (const float*)d_in[5];
    const float* dt_bias      = (const float*)d_in[6];
    const float* A_log        = (const float*)d_in[7];
    const float* Dvec         = (const float*)d_in[8];
    const float* norm_w       = (const float*)d_in[9];
    const float* out_proj_w   = (const float*)d_in[10];
    float* out = (float*)d_out;

    float* ws = (float*)d_ws;
    size_t o = 0;
    float* proj = ws + o; o += (size_t)BATCH * PROJ_DIM;
    float* hs   = ws + o; o += (size_t)BATCH * INTER;
    float* cm   = ws + o; o += (size_t)BATCH * NSTATE;
    float* dts  = ws + o; o += (size_t)BATCH * NH;
    float* da   = ws + o; o += (size_t)BATCH * NH;
    float* bc   = ws + o; o += (size_t)BATCH;
    float* y    = ws + o; o += (size_t)BATCH * INTER;
    float* yn   = ws + o; o += (size_t)BATCH * INTER;
    (void)o; (void)ws_size; (void)in_sizes; (void)n_in;

    // zero accumulation targets (graph-capturable memset nodes)
    hipMemsetAsync(proj, 0, (size_t)BATCH * PROJ_DIM * sizeof(float), stream);
    hipMemsetAsync(out, 0, (size_t)out_size * sizeof(float), stream);

    // 1) in_proj: N=10576 (661 tiles), K=2560, split-K=4 -> 2644 waves
    {
        const int ksplit = 4;
        const int waves  = (PROJ_DIM / 16) * ksplit;
        const int blocks = (waves + 7) / 8;
        gemm_nt_f32_wmma<<<blocks, 256, 0, stream>>>(
            input_states, in_proj_w, proj, PROJ_DIM, HID, ksplit);
    }

    // 2) conv update + SiLU + dt/dA + <Bm,Cm>
    conv_dt_kernel<<<BATCH, 256, 0, stream>>>(
        proj, conv_state, conv_w, conv_b, dt_bias, A_log, hs, cm, dts, da, bc);

    // 3) SSM contraction (168 MB stream)
    ssm_kernel<<<dim3(NH, BATCH), 256, 0, stream>>>(
        ssm_state, cm, hs, dts, da, bc, Dvec, y);

    // 4) gate + RMS norm
    gatenorm_kernel<<<BATCH, 256, 0, stream>>>(proj, y, norm_w, yn);

    // 5) out_proj: N=2560 (160 tiles), K=5120, split-K=8 -> 1280 waves
    {
        const int ksplit = 8;
        const int waves  = (HID / 16) * ksplit;
        const int blocks = (waves + 7) / 8;
        gemm_nt_f32_wmma<<<blocks, 256, 0, stream>>>(
            yn, out_proj_w, out, HID, INTER, ksplit);
    }
}